// LAAttention_11673721111010
// MI455X (gfx1250) — compile-verified
//
#include <hip/hip_runtime.h>

// Problem constants (from reference setup_inputs)
#define BDIM 2
#define CDIM 128
#define CQK  32
#define NDIM 8192
#define SDIM 16
#define BN_EPS 1e-5f

typedef float v2f __attribute__((ext_vector_type(2)));
typedef float v8f __attribute__((ext_vector_type(8)));

// D(16x16 f32) += A(16x4 f32) * B(4x16 f32)
__device__ __forceinline__ v8f wmma4(v2f a, v2f b, v8f c) {
  return __builtin_amdgcn_wmma_f32_16x16x4_f32(false, a, false, b, (short)0, c,
                                               false, false);
}

// ---------------------------------------------------------------------------
// Kernel 1: q/k/v 1x1-conv projections as WMMA GEMMs.
// grid = (N/16, 12, B); one wave (32 threads) per 16x16 output tile.
// task 0..7  -> v (C=128, 8 M-tiles)
// task 8..9  -> k (Cqk=32, 2 M-tiles)
// task 10..11-> q (Cqk=32, 2 M-tiles)
// A-frag (16x4 f32): lane L holds W[mtile*16 + (L&15)][k0 + 2*(L>>4) + {0,1}]
// B-frag (4x16 f32): lane L holds Act[k0 + 2*(L>>4) + {0,1}][ntile*16 + (L&15)]
// D (16x16 f32):     vgpr i, half h -> row mtile*16 + i + 8h, col = L&15
// ---------------------------------------------------------------------------
__global__ __launch_bounds__(32)
void proj_gemm(const float* __restrict__ x, const float* __restrict__ y,
               const float* __restrict__ wq, const float* __restrict__ bq,
               const float* __restrict__ wk, const float* __restrict__ bk,
               const float* __restrict__ wv, const float* __restrict__ bv,
               float* __restrict__ q_ws, float* __restrict__ k_ws,
               float* __restrict__ v_ws) {
  const int lane = threadIdx.x;
  const int col  = lane & 15;
  const int half = lane >> 4;
  const int task = blockIdx.y;
  const int b    = blockIdx.z;
  const int ncol = blockIdx.x * 16 + col;

  const float* W; const float* bias; const float* act; float* out;
  int mtile, Odim;
  if (task < 8)       { W = wv; bias = bv; act = y; out = v_ws; mtile = task;      Odim = CDIM; }
  else if (task < 10) { W = wk; bias = bk; act = y; out = k_ws; mtile = task - 8;  Odim = CQK;  }
  else                { W = wq; bias = bq; act = x; out = q_ws; mtile = task - 10; Odim = CQK;  }

  const float* actb = act + (size_t)b * CDIM * NDIM;
  const int m = mtile * 16 + col;

  v8f acc = {};
#pragma unroll
  for (int t = 0; t < 32; ++t) {
    const int k0 = 4 * t;
    v2f a, bf;
    a.x  = W[m * CDIM + k0 + 2 * half + 0];
    a.y  = W[m * CDIM + k0 + 2 * half + 1];
    bf.x = actb[(size_t)(k0 + 2 * half + 0) * NDIM + ncol];
    bf.y = actb[(size_t)(k0 + 2 * half + 1) * NDIM + ncol];
    acc = wmma4(a, bf, acc);
  }

  float* outb = out + (size_t)b * Odim * NDIM;
#pragma unroll
  for (int i = 0; i < 8; ++i) {
    const int row = mtile * 16 + i + 8 * half;
    outb[(size_t)row * NDIM + ncol] = acc[i] + bias[row];
  }
}

// ---------------------------------------------------------------------------
// Kernel 2: per-point neighborhood attention. One wave per point (b, n).
// Lane: col = neighbor s (0..15), half = lane>>4.
// LDS tile sT[32][16] restages C-layout results into B-fragments.
// ---------------------------------------------------------------------------
__global__ __launch_bounds__(32)
void neigh_attn(const float* __restrict__ y_xyz, const int* __restrict__ idx,
                const float* __restrict__ q_ws, const float* __restrict__ k_ws,
                const float* __restrict__ v_ws,
                const float* __restrict__ gw1, const float* __restrict__ gb1,
                const float* __restrict__ g_gamma, const float* __restrict__ g_beta,
                const float* __restrict__ g_mean, const float* __restrict__ g_var,
                const float* __restrict__ gw2, const float* __restrict__ gb2,
                const float* __restrict__ n1_gamma, const float* __restrict__ n1_beta,
                const float* __restrict__ n1_mean, const float* __restrict__ n1_var,
                const float* __restrict__ nw1, const float* __restrict__ nb1,
                const float* __restrict__ n2_gamma, const float* __restrict__ n2_beta,
                const float* __restrict__ n2_mean, const float* __restrict__ n2_var,
                const float* __restrict__ nw2, const float* __restrict__ nb2,
                float* __restrict__ out) {
  __shared__ float sT[CQK * SDIM];  // 32 rows x 16 cols staging tile

  const int lane = threadIdx.x;
  const int col  = lane & 15;
  const int half = lane >> 4;
  const int pid  = blockIdx.x;      // 0 .. B*N-1
  const int b    = pid >> 13;       // N = 8192
  const int n    = pid & (NDIM - 1);

  // neighbor index for this lane's column (same in both lane halves)
  const int j = idx[((size_t)b * NDIM + n) * SDIM + col];

  // -------- Stage A: geometry encoding enc[4][16] kept directly as B-frag
  const float cx = y_xyz[((size_t)b * 3 + 0) * NDIM + n];
  const float cy = y_xyz[((size_t)b * 3 + 1) * NDIM + n];
  const float cz = y_xyz[((size_t)b * 3 + 2) * NDIM + n];
  const float dx = y_xyz[((size_t)b * 3 + 0) * NDIM + j] - cx;
  const float dy = y_xyz[((size_t)b * 3 + 1) * NDIM + j] - cy;
  const float dz = y_xyz[((size_t)b * 3 + 2) * NDIM + j] - cz;
  const float dist = sqrtf(dx * dx + dy * dy + dz * dz);
  const float e[4] = {dx, dy, dz, dist};
  v2f encB;
  encB.x = e[2 * half + 0];
  encB.y = e[2 * half + 1];

  // -------- Stage B: h1 = relu(bn_g(gw1 @ enc + gb1)) -> LDS
#pragma unroll
  for (int mt = 0; mt < 2; ++mt) {
    const int m = mt * 16 + col;
    v2f a;
    a.x = gw1[m * 4 + 2 * half + 0];
    a.y = gw1[m * 4 + 2 * half + 1];
    v8f c = {};
    c = wmma4(a, encB, c);
#pragma unroll
    for (int i = 0; i < 8; ++i) {
      const int r = mt * 16 + i + 8 * half;
      const float inv = g_gamma[r] * rsqrtf(g_var[r] + BN_EPS);
      float val = (c[i] + gb1[r]) * inv + (g_beta[r] - g_mean[r] * inv);
      sT[r * 16 + col] = fmaxf(val, 0.f);
    }
  }

  // -------- Stage C: h = gw2 @ h1 (keep in registers; bias added in stage D)
  v8f hacc[2] = {};
#pragma unroll
  for (int mt = 0; mt < 2; ++mt) {
    const int m = mt * 16 + col;
#pragma unroll
    for (int t = 0; t < 8; ++t) {
      const int k0 = 4 * t;
      v2f a, bf;
      a.x  = gw2[m * CQK + k0 + 2 * half + 0];
      a.y  = gw2[m * CQK + k0 + 2 * half + 1];
      bf.x = sT[(k0 + 2 * half + 0) * 16 + col];
      bf.y = sT[(k0 + 2 * half + 1) * 16 + col];
      hacc[mt] = wmma4(a, bf, hacc[mt]);
    }
  }

  // -------- Stage D: emb = -(k-q)^2 + h + gb2; relu(bn_n1(emb)) -> LDS
#pragma unroll
  for (int mt = 0; mt < 2; ++mt) {
#pragma unroll
    for (int i = 0; i < 8; ++i) {
      const int r  = mt * 16 + i + 8 * half;
      const float qv = q_ws[((size_t)b * CQK + r) * NDIM + n];
      const float kv = k_ws[((size_t)b * CQK + r) * NDIM + j];
      const float df = kv - qv;
      float val = hacc[mt][i] + gb2[r] - df * df;
      const float inv = n1_gamma[r] * rsqrtf(n1_var[r] + BN_EPS);
      val = val * inv + (n1_beta[r] - n1_mean[r] * inv);
      sT[r * 16 + col] = fmaxf(val, 0.f);
    }
  }

  // -------- Stage E: t2 = relu(bn_n2(nw1 @ t1 + nb1)) -> LDS
  v8f tacc[2] = {};
#pragma unroll
  for (int mt = 0; mt < 2; ++mt) {
    const int m = mt * 16 + col;
#pragma unroll
    for (int t = 0; t < 8; ++t) {
      const int k0 = 4 * t;
      v2f a, bf;
      a.x  = nw1[m * CQK + k0 + 2 * half + 0];
      a.y  = nw1[m * CQK + k0 + 2 * half + 1];
      bf.x = sT[(k0 + 2 * half + 0) * 16 + col];
      bf.y = sT[(k0 + 2 * half + 1) * 16 + col];
      tacc[mt] = wmma4(a, bf, tacc[mt]);
    }
  }
#pragma unroll
  for (int mt = 0; mt < 2; ++mt) {
#pragma unroll
    for (int i = 0; i < 8; ++i) {
      const int r = mt * 16 + i + 8 * half;
      float val = tacc[mt][i] + nb1[r];
      const float inv = n2_gamma[r] * rsqrtf(n2_var[r] + BN_EPS);
      val = val * inv + (n2_beta[r] - n2_mean[r] * inv);
      sT[r * 16 + col] = fmaxf(val, 0.f);
    }
  }

  // -------- Stage F: w = nw2 @ t2 + nb2 ; softmax over S ; out = sum_s v*w
  for (int mt = 0; mt < 8; ++mt) {
    const int m = mt * 16 + col;
    v8f w = {};
#pragma unroll
    for (int t = 0; t < 8; ++t) {
      const int k0 = 4 * t;
      v2f a, bf;
      a.x  = nw2[m * CQK + k0 + 2 * half + 0];
      a.y  = nw2[m * CQK + k0 + 2 * half + 1];
      bf.x = sT[(k0 + 2 * half + 0) * 16 + col];
      bf.y = sT[(k0 + 2 * half + 1) * 16 + col];
      w = wmma4(a, bf, w);
    }
#pragma unroll
    for (int i = 0; i < 8; ++i) {
      const int r = mt * 16 + i + 8 * half;  // channel in [0,128)
      const float wl = w[i] + nb2[r];
      // softmax across the 16 neighbor columns: each row's 16 values live in
      // one 16-lane half, so xor-shuffles 1/2/4/8 stay within the row.
      float mx = wl;
      mx = fmaxf(mx, __shfl_xor(mx, 1));
      mx = fmaxf(mx, __shfl_xor(mx, 2));
      mx = fmaxf(mx, __shfl_xor(mx, 4));
      mx = fmaxf(mx, __shfl_xor(mx, 8));
      const float ex = __expf(wl - mx);
      float sm = ex;
      sm += __shfl_xor(sm, 1);
      sm += __shfl_xor(sm, 2);
      sm += __shfl_xor(sm, 4);
      sm += __shfl_xor(sm, 8);
      const float prob = ex / sm;
      // gather v (L2-resident: v_ws is 8 MB << 192 MB L2) and reduce
      float o = prob * v_ws[((size_t)b * CDIM + r) * NDIM + j];
      o += __shfl_xor(o, 1);
      o += __shfl_xor(o, 2);
      o += __shfl_xor(o, 4);
      o += __shfl_xor(o, 8);
      if (col == 0) out[((size_t)b * CDIM + r) * NDIM + n] = o;
    }
  }
}

// ---------------------------------------------------------------------------
extern "C" void kernel_launch(void* const* d_in, const int* in_sizes, int n_in,
                              void* d_out, int out_size, void* d_ws, size_t ws_size,
                              hipStream_t stream) {
  const float* x      = (const float*)d_in[0];
  const float* y      = (const float*)d_in[1];
  const float* y_xyz  = (const float*)d_in[2];
  const int*   idx    = (const int*)d_in[3];
  const float* wq     = (const float*)d_in[4];
  const float* bq     = (const float*)d_in[5];
  const float* wk     = (const float*)d_in[6];
  const float* bk     = (const float*)d_in[7];
  const float* wv     = (const float*)d_in[8];
  const float* bv     = (const float*)d_in[9];
  const float* gw1    = (const float*)d_in[10];
  const float* gb1    = (const float*)d_in[11];
  const float* g_gamma= (const float*)d_in[12];
  const float* g_beta = (const float*)d_in[13];
  const float* g_mean = (const float*)d_in[14];
  const float* g_var  = (const float*)d_in[15];
  const float* gw2    = (const float*)d_in[16];
  const float* gb2    = (const float*)d_in[17];
  const float* n1_gamma = (const float*)d_in[18];
  const float* n1_beta  = (const float*)d_in[19];
  const float* n1_mean  = (const float*)d_in[20];
  const float* n1_var   = (const float*)d_in[21];
  const float* nw1    = (const float*)d_in[22];
  const float* nb1    = (const float*)d_in[23];
  const float* n2_gamma = (const float*)d_in[24];
  const float* n2_beta  = (const float*)d_in[25];
  const float* n2_mean  = (const float*)d_in[26];
  const float* n2_var   = (const float*)d_in[27];
  const float* nw2    = (const float*)d_in[28];
  const float* nb2    = (const float*)d_in[29];

  // workspace: q [B,32,N] | k [B,32,N] | v [B,128,N]  (12 MB total)
  float* q_ws = (float*)d_ws;
  float* k_ws = q_ws + (size_t)BDIM * CQK * NDIM;
  float* v_ws = k_ws + (size_t)BDIM * CQK * NDIM;

  dim3 g1(NDIM / 16, 12, BDIM);
  proj_gemm<<<g1, 32, 0, stream>>>(x, y, wq, bq, wk, bk, wv, bv,
                                   q_ws, k_ws, v_ws);

  dim3 g2(BDIM * NDIM, 1, 1);
  neigh_attn<<<g2, 32, 0, stream>>>(y_xyz, idx, q_ws, k_ws, v_ws,
                                    gw1, gb1, g_gamma, g_beta, g_mean, g_var,
                                    gw2, gb2,
                                    n1_gamma, n1_beta, n1_mean, n1_var,
                                    nw1, nb1,
                                    n2_gamma, n2_beta, n2_mean, n2_var,
                                    nw2, nb2, (float*)d_out);
}